// DynPredNet_84069689852207
// MI455X (gfx1250) — compile-verified
//
#include <hip/hip_runtime.h>
#include <math.h>

// ---------------- CDNA5 WMMA f32 16x16x4 ----------------
typedef float v2f __attribute__((ext_vector_type(2)));
typedef float v8f __attribute__((ext_vector_type(8)));

__device__ __forceinline__ v8f wmma4(v2f a, v2f b, v8f c) {
  // (neg_a, A, neg_b, B, c_mod, C, reuse_a, reuse_b)
  return __builtin_amdgcn_wmma_f32_16x16x4_f32(false, a, false, b, (short)0, c, false, false);
}

// ---- async global->LDS staging of the per-block B panel (CDNA5 TDM-lite path) ----
// abT panel: 16 rows (n0..n0+15) x K floats, LDS row stride K+4 (bank-conflict pad).
__device__ __forceinline__ void stage_T(const float* Bsrc, int K, float* Bl) {
  int kc = K >> 2;                         // b128 chunks per row
  int row = threadIdx.x >> 4;              // 16 threads per row
  int q0  = threadIdx.x & 15;
  for (int q = q0; q < kc; q += 16) {
    unsigned lds = (unsigned)(unsigned long long)(Bl + row * (K + 4) + q * 4);
    const float* g = Bsrc + row * K + q * 4;
    asm volatile("global_load_async_to_lds_b128 %0, %1, off" :: "v"(lds), "v"(g) : "memory");
  }
  asm volatile("s_wait_asynccnt 0x0" ::: "memory");
  __syncthreads();
}

// abN panel: K rows x 16 cols (n0..n0+15), LDS layout [K][16].
__device__ __forceinline__ void stage_N(const float* Bsrc, int K, int N, int n0, float* Bl) {
  int q = threadIdx.x & 3;                 // 4 b128 chunks per row
  for (int row = (int)(threadIdx.x >> 2); row < K; row += 64) {
    unsigned lds = (unsigned)(unsigned long long)(Bl + row * 16 + q * 4);
    const float* g = Bsrc + row * N + n0 + q * 4;
    asm volatile("global_load_async_to_lds_b128 %0, %1, off" :: "v"(lds), "v"(g) : "memory");
  }
  asm volatile("s_wait_asynccnt 0x0" ::: "memory");
  __syncthreads();
}

// Tile D[16,16] = A[16,K] * panel^T, A from global, B panel from LDS (stride K+4)
__device__ __forceinline__ v8f tile_aG_bT_lds(const float* A, const float* Bl, int K,
                                              int m0, int lane) {
  int r = lane & 15, hi = lane >> 4;
  const float* ap = A + (m0 + r) * K + hi * 2;
  const float* bp = Bl + r * (K + 4) + hi * 2;
  v8f acc = {0.f,0.f,0.f,0.f,0.f,0.f,0.f,0.f};
  for (int k = 0; k < K; k += 4) {
    v2f a; a.x = ap[k]; a.y = ap[k + 1];
    v2f b; b.x = bp[k]; b.y = bp[k + 1];
    acc = wmma4(a, b, acc);
  }
  return acc;
}

// Tile D[16,16] = A[16,K] * panel, A from global, B panel from LDS ([K][16])
__device__ __forceinline__ v8f tile_aG_bN_lds(const float* A, const float* Bl, int K,
                                              int m0, int lane) {
  int r = lane & 15, hi = lane >> 4;
  const float* ap = A + (m0 + r) * K + hi * 2;
  const float* bp = Bl + (hi * 2) * 16 + r;
  v8f acc = {0.f,0.f,0.f,0.f,0.f,0.f,0.f,0.f};
  for (int k = 0; k < K; k += 4) {
    v2f a; a.x = ap[k]; a.y = ap[k + 1];
    v2f b; b.x = bp[k * 16]; b.y = bp[(k + 1) * 16];
    acc = wmma4(a, b, acc);
  }
  return acc;
}

// ---------------- constants ----------------
#define BATCH 128
#define RD 256
#define R2D 128
#define MIXD 32
#define HIDD 512
#define IND 1024
#define LR_R 0.01f
#define LMDA_R 0.01f
#define LR_R2 0.001f
#define LMDA_R2 1e-4f
#define C0 (2.0f / 128.0f)

// dynamic LDS sizes for staged panels
#define SMT(K) ((size_t)(16 * ((K) + 4) * 4))
#define SMN(K) ((size_t)((K) * 16 * 4))

// ---------------- kernels ----------------

// C[128,N] = A[128,K] @ B[N,K]^T + bias ; grid = N/16 blocks, 256 thr (8 waves = 8 m-tiles)
__global__ void k_gemm_abT(const float* __restrict__ A, const float* __restrict__ B,
                           const float* __restrict__ bias, float* __restrict__ C,
                           int K, int N, const int* gate) {
  if (gate[0]) return;
  extern __shared__ float Bl[];
  int lane = threadIdx.x & 31, wid = threadIdx.x >> 5;
  int m0 = wid * 16, n0 = blockIdx.x * 16;
  stage_T(B + n0 * K, K, Bl);
  v8f acc = tile_aG_bT_lds(A, Bl, K, m0, lane);
  int r = lane & 15, hi = lane >> 4;
  float bv = bias ? bias[n0 + r] : 0.f;
  for (int v = 0; v < 8; v++)
    C[(m0 + v + 8 * hi) * N + n0 + r] = acc[v] + bv;
}

// C[128,N] = A[128,K] @ B[K,N]
__global__ void k_gemm_abN(const float* __restrict__ A, const float* __restrict__ B,
                           float* __restrict__ C, int K, int N, const int* gate) {
  if (gate[0]) return;
  extern __shared__ float Bl[];
  int lane = threadIdx.x & 31, wid = threadIdx.x >> 5;
  int m0 = wid * 16, n0 = blockIdx.x * 16;
  stage_N(B, K, N, n0, Bl);
  v8f acc = tile_aG_bN_lds(A, Bl, K, m0, lane);
  int r = lane & 15, hi = lane >> 4;
  for (int v = 0; v < 8; v++)
    C[(m0 + v + 8 * hi) * N + n0 + r] = acc[v];
}

// U[b,m,i] = sum_j temporal[m,i,j] * r_prev[b,j]; grid (16, 32)
__global__ void k_gemm_U(const float* __restrict__ rp, const float* __restrict__ Tm,
                         float* __restrict__ U) {
  extern __shared__ float Bl[];
  int lane = threadIdx.x & 31, wid = threadIdx.x >> 5;
  int m = blockIdx.y;
  int ni = blockIdx.x, mi = wid;          // all waves share ni -> shared B panel
  int n0 = ni * 16;
  stage_T(Tm + m * (RD * RD) + n0 * RD, RD, Bl);
  v8f acc = tile_aG_bT_lds(rp, Bl, RD, mi * 16, lane);
  int r = lane & 15, hi = lane >> 4;
  for (int v = 0; v < 8; v++) {
    int b = mi * 16 + v + 8 * hi, i = n0 + r;
    U[b * (MIXD * RD) + m * RD + i] = acc[v];
  }
}

// e = r @ Wd^T - x  ; Wd[1024,256], x row stride 6144; grid 64 blocks
__global__ void k_sperr(const float* __restrict__ rcur, const float* __restrict__ Wd,
                        const float* __restrict__ X, int xstride,
                        float* __restrict__ e, const int* gate) {
  if (gate[0]) return;
  extern __shared__ float Bl[];
  int lane = threadIdx.x & 31, wid = threadIdx.x >> 5;
  int m0 = wid * 16, n0 = blockIdx.x * 16;
  stage_T(Wd + n0 * RD, RD, Bl);
  v8f acc = tile_aG_bT_lds(rcur, Bl, RD, m0, lane);
  int r = lane & 15, hi = lane >> 4;
  for (int v = 0; v < 8; v++) {
    int m = m0 + v + 8 * hi, n = n0 + r;
    e[m * IND + n] = acc[v] - X[m * xstride + n];
  }
}

// fused: gsp = e @ Wd (abN, K=1024, N=256); r = max(r - LR*(c0*gsp + c0*(r-pred)) - l, 0)
// + block partial sums of ||rn-r||^2, ||r||^2 ; grid 16 blocks
__global__ void k_rupd(const float* __restrict__ e, const float* __restrict__ Wd,
                       const float* __restrict__ pred, int use_temp,
                       float* __restrict__ rbuf, float* __restrict__ parts,
                       const int* gate) {
  if (gate[0]) return;
  extern __shared__ float Bl[];
  int lane = threadIdx.x & 31, wid = threadIdx.x >> 5;
  int m0 = wid * 16, n0 = blockIdx.x * 16;
  stage_N(Wd, IND, RD, n0, Bl);
  v8f acc = tile_aG_bN_lds(e, Bl, IND, m0, lane);
  int r = lane & 15, hi = lane >> 4;
  float dd = 0.f, nn = 0.f;
  for (int v = 0; v < 8; v++) {
    int idx = (m0 + v + 8 * hi) * RD + n0 + r;
    float ro = rbuf[idx];
    float gr = C0 * acc[v];
    if (use_temp) gr += C0 * (ro - pred[idx]);
    float y = ro - LR_R * gr - LMDA_R;
    float rn = y > 0.f ? y : 0.f;       // relu(relu(y-l)-relu(-y-l)) == max(y-l,0)
    float d = rn - ro;
    dd += d * d; nn += ro * ro;
    rbuf[idx] = rn;
  }
  __shared__ float sd[256], sn[256];
  sd[threadIdx.x] = dd; sn[threadIdx.x] = nn;
  __syncthreads();
  for (int s = 128; s > 0; s >>= 1) {
    if (threadIdx.x < s) { sd[threadIdx.x] += sd[threadIdx.x + s]; sn[threadIdx.x] += sn[threadIdx.x + s]; }
    __syncthreads();
  }
  if (threadIdx.x == 0) { parts[blockIdx.x * 2] = sd[0]; parts[blockIdx.x * 2 + 1] = sn[0]; }
}

// LayerNorm+ELU fwd: wave per row of [128,512]; saves h1, xhat, rinv; grid 16 blocks
__global__ void k_ln_fwd(const float* __restrict__ z1, const float* __restrict__ g,
                         const float* __restrict__ be, float* __restrict__ h1,
                         float* __restrict__ xhat, float* __restrict__ rinv,
                         const int* gate) {
  if (gate[0]) return;
  int wid = threadIdx.x >> 5, lane = threadIdx.x & 31;
  int row = blockIdx.x * 8 + wid;
  const float* zr = z1 + row * HIDD;
  float vals[16];
  float s = 0.f;
  for (int j = 0; j < 16; j++) { vals[j] = zr[lane + j * 32]; s += vals[j]; }
  for (int o = 16; o > 0; o >>= 1) s += __shfl_xor(s, o, 32);
  float mu = s * (1.0f / HIDD);
  float vs = 0.f;
  for (int j = 0; j < 16; j++) { float d = vals[j] - mu; vs += d * d; }
  for (int o = 16; o > 0; o >>= 1) vs += __shfl_xor(vs, o, 32);
  float rv = rsqrtf(vs * (1.0f / HIDD) + 1e-5f);
  if (lane == 0) rinv[row] = rv;
  for (int j = 0; j < 16; j++) {
    int c = lane + j * 32;
    float xh = (vals[j] - mu) * rv;
    float a = xh * g[c] + be[c];
    float h = a > 0.f ? a : expm1f(a);
    xhat[row * HIDD + c] = xh;
    h1[row * HIDD + c] = h;
  }
}

// LayerNorm+ELU bwd: dz1 from dh1; grid 16 blocks
__global__ void k_ln_bwd(const float* __restrict__ dh1, const float* __restrict__ h1,
                         const float* __restrict__ xhat, const float* __restrict__ rinv,
                         const float* __restrict__ g, float* __restrict__ dz1,
                         const int* gate) {
  if (gate[0]) return;
  int wid = threadIdx.x >> 5, lane = threadIdx.x & 31;
  int row = blockIdx.x * 8 + wid;
  const float* dhr = dh1 + row * HIDD;
  const float* hr  = h1  + row * HIDD;
  const float* xr  = xhat + row * HIDD;
  float dx[16], xh[16];
  float s1 = 0.f, s2 = 0.f;
  for (int j = 0; j < 16; j++) {
    int c = lane + j * 32;
    float h = hr[c];
    float d = dhr[c] * (h > 0.f ? 1.f : (h + 1.f)) * g[c];  // elu' = exp(a) = h+1 (a<=0)
    dx[j] = d; xh[j] = xr[c];
    s1 += d; s2 += d * xh[j];
  }
  for (int o = 16; o > 0; o >>= 1) { s1 += __shfl_xor(s1, o, 32); s2 += __shfl_xor(s2, o, 32); }
  s1 *= (1.0f / HIDD); s2 *= (1.0f / HIDD);
  float rv = rinv[row];
  for (int j = 0; j < 16; j++) {
    int c = lane + j * 32;
    dz1[row * HIDD + c] = rv * (dx[j] - s1 - xh[j] * s2);
  }
}

// pred[b,i] = relu(sum_m w[b,m] * U[b,m,i]); grid 128 blocks x 256 thr
__global__ void k_pred(const float* __restrict__ U, const float* __restrict__ wv,
                       float* __restrict__ pred, float* __restrict__ pred0,
                       int save0, const int* gate) {
  if (gate[0]) return;
  int b = blockIdx.x, i = threadIdx.x;
  const float* Ub = U + b * (MIXD * RD) + i;
  const float* wb = wv + b * MIXD;
  float s = 0.f;
  for (int m = 0; m < MIXD; m++) s += wb[m] * Ub[m * RD];
  float p = s > 0.f ? s : 0.f;
  pred[b * RD + i] = p;
  if (save0) pred0[b * RD + i] = p;
}

// dpre = (pred>0) ? (2/B)*(pred - r) : 0 ; grid 128x256
__global__ void k_dpre(const float* __restrict__ pred, const float* __restrict__ r,
                       float* __restrict__ dpre, const int* gate) {
  if (gate[0]) return;
  int idx = blockIdx.x * 256 + threadIdx.x;
  float p = pred[idx];
  dpre[idx] = p > 0.f ? C0 * (p - r[idx]) : 0.f;
}

// dw[b,m] = sum_i dpre[b,i]*U[b,m,i]; wave per (b,m); grid 512 blocks
__global__ void k_dw(const float* __restrict__ dpre, const float* __restrict__ U,
                     float* __restrict__ dw, const int* gate) {
  if (gate[0]) return;
  int wid = threadIdx.x >> 5, lane = threadIdx.x & 31;
  int wg = blockIdx.x * 8 + wid;
  int b = wg >> 5, m = wg & 31;
  const float* dp = dpre + b * RD;
  const float* Ub = U + b * (MIXD * RD) + m * RD;
  float s = 0.f;
  for (int i = lane; i < RD; i += 32) s += dp[i] * Ub[i];
  for (int o = 16; o > 0; o >>= 1) s += __shfl_xor(s, o, 32);
  if (lane == 0) dw[b * MIXD + m] = s;
}

// Adam on r2 (in place) + convergence partials; grid 64 blocks
__global__ void k_adam(const float* __restrict__ dr2g, float* __restrict__ r2,
                       float* __restrict__ ma, float* __restrict__ va,
                       float bc1, float bc2, float* __restrict__ parts, const int* gate) {
  if (gate[0]) return;
  int idx = blockIdx.x * 256 + threadIdx.x;
  float r2o = r2[idx];
  float g2 = dr2g[idx] + LMDA_R2 * r2o;
  float m = 0.9f * ma[idx] + (1.0f - 0.9f) * g2;
  float v = 0.999f * va[idx] + (1.0f - 0.999f) * g2 * g2;
  ma[idx] = m; va[idx] = v;
  float r2n = r2o - LR_R2 * (m * bc1) / (sqrtf(v * bc2) + 1e-8f);
  r2[idx] = r2n;
  float d = r2n - r2o;
  __shared__ float sd[256], sn[256];
  sd[threadIdx.x] = d * d; sn[threadIdx.x] = r2o * r2o;
  __syncthreads();
  for (int s = 128; s > 0; s >>= 1) {
    if (threadIdx.x < s) { sd[threadIdx.x] += sd[threadIdx.x + s]; sn[threadIdx.x] += sn[threadIdx.x + s]; }
    __syncthreads();
  }
  if (threadIdx.x == 0) { parts[blockIdx.x * 2] = sd[0]; parts[blockIdx.x * 2 + 1] = sn[0]; }
}

// convergence finalize: flags[next] = flags[prev] | conv
__global__ void k_fin(const float* __restrict__ pr, int nr, const float* __restrict__ p2,
                      int n2, int use2, int* __restrict__ flags, int sprev, int snext) {
  if (threadIdx.x == 0) {
    float dr = 0.f, rr = 0.f;
    for (int i = 0; i < nr; i++) { dr += pr[2 * i]; rr += pr[2 * i + 1]; }
    int conv = (sqrtf(dr) / (sqrtf(rr) + 1e-16f)) < 1e-3f;
    if (use2) {
      float d2 = 0.f, r2s = 0.f;
      for (int i = 0; i < n2; i++) { d2 += p2[2 * i]; r2s += p2[2 * i + 1]; }
      conv = conv && ((sqrtf(d2) / (sqrtf(r2s) + 1e-16f)) < 1e-3f);
    }
    flags[snext] = flags[sprev] | conv;
  }
}

// parts[blk] = block sum of (a-b)^2 (b may be null -> a^2); grid n/256 blocks
__global__ void k_sqred(const float* __restrict__ a, const float* __restrict__ b,
                        float* __restrict__ parts) {
  __shared__ float sm[256];
  int idx = blockIdx.x * 256 + threadIdx.x;
  float x = b ? (a[idx] - b[idx]) : a[idx];
  sm[threadIdx.x] = x * x;
  __syncthreads();
  for (int s = 128; s > 0; s >>= 1) {
    if (threadIdx.x < s) sm[threadIdx.x] += sm[threadIdx.x + s];
    __syncthreads();
  }
  if (threadIdx.x == 0) parts[blockIdx.x] = sm[0];
}

__global__ void k_addout(float* __restrict__ out, const float* __restrict__ parts,
                         int n, float scale) {
  if (threadIdx.x == 0) {
    float s = 0.f;
    for (int i = 0; i < n; i++) s += parts[i];
    out[0] += scale * s;
  }
}

// r2_losses[b, tcol] = sum_i (r[b,i]-pred0[b,i])^2; wave per row; grid 16 blocks
__global__ void k_r2loss(const float* __restrict__ r, const float* __restrict__ pred0,
                         float* __restrict__ out2, int tcol) {
  int wid = threadIdx.x >> 5, lane = threadIdx.x & 31;
  int b = blockIdx.x * 8 + wid;
  const float* rr = r + b * RD;
  const float* pp = pred0 + b * RD;
  float s = 0.f;
  for (int i = lane; i < RD; i += 32) { float d = rr[i] - pp[i]; s += d * d; }
  for (int o = 16; o > 0; o >>= 1) s += __shfl_xor(s, o, 32);
  if (lane == 0) out2[b * 5 + tcol] = s;
}

__global__ void k_copy(float* __restrict__ dst, const float* __restrict__ src) {
  int idx = blockIdx.x * 256 + threadIdx.x;
  dst[idx] = src[idx];
}

__global__ void k_init(float* __restrict__ rcur, float* __restrict__ r2,
                       float* __restrict__ dout, int* __restrict__ flags) {
  int idx = blockIdx.x * 256 + threadIdx.x;   // 32768 threads
  rcur[idx] = 0.f;
  if (idx < 16384) r2[idx] = 0.f;
  if (idx < 128) flags[idx] = 0;
  if (idx < 2) dout[idx] = 0.f;
}

__global__ void k_stepinit(float* __restrict__ rprev, float* __restrict__ rcur,
                           float* __restrict__ ma, float* __restrict__ va) {
  int idx = blockIdx.x * 256 + threadIdx.x;   // 32768 threads
  rprev[idx] = rcur[idx];
  rcur[idx] = 0.f;
  if (idx < 16384) { ma[idx] = 0.f; va[idx] = 0.f; }
}

// ---------------- workspace layout (floats) ----------------
#define OFF_E       0            // 131072
#define OFF_U       131072       // 1048576
#define OFF_Z1      1179648      // 65536
#define OFF_H1      1245184
#define OFF_XHAT    1310720
#define OFF_Z2      1376256
#define OFF_DZ2     1441792
#define OFF_DH1     1507328
#define OFF_DZ1     1572864
#define OFF_RINV    1638400      // 128
#define OFF_W       1638528      // 4096
#define OFF_DW      1642624      // 4096
#define OFF_PRED    1646720      // 32768
#define OFF_PRED0   1679488
#define OFF_DPRE    1712256
#define OFF_RCUR    1745024
#define OFF_RPREV   1777792
#define OFF_R2      1810560      // 16384
#define OFF_MA      1826944
#define OFF_VA      1843328
#define OFF_DR2G    1859712
#define OFF_PARTSR  1876096      // 32
#define OFF_PARTSR2 1876128      // 128
#define OFF_PARTSB  1876256      // 512
#define OFF_FLAGS   1876768      // 128 ints

extern "C" void kernel_launch(void* const* d_in, const int* in_sizes, int n_in,
                              void* d_out, int out_size, void* d_ws, size_t ws_size,
                              hipStream_t stream) {
  (void)in_sizes; (void)n_in; (void)out_size; (void)ws_size;
  const float* X  = (const float*)d_in[0];   // [128,6,1024]
  const float* Wd = (const float*)d_in[1];   // [1024,256]
  const float* Tm = (const float*)d_in[2];   // [32,256,256]
  const float* W1 = (const float*)d_in[3];   // [512,128]
  const float* b1 = (const float*)d_in[4];
  const float* gg = (const float*)d_in[5];
  const float* be = (const float*)d_in[6];
  const float* W2 = (const float*)d_in[7];   // [512,512]
  const float* b2 = (const float*)d_in[8];
  const float* W3 = (const float*)d_in[9];   // [32,512]
  const float* b3 = (const float*)d_in[10];
  float* out = (float*)d_out;                // [2 + 640 + 32768 + 16384]
  float* ws  = (float*)d_ws;

  float* e     = ws + OFF_E;
  float* U     = ws + OFF_U;
  float* z1    = ws + OFF_Z1;
  float* h1    = ws + OFF_H1;
  float* xhat  = ws + OFF_XHAT;
  float* z2    = ws + OFF_Z2;
  float* dz2   = ws + OFF_DZ2;
  float* dh1   = ws + OFF_DH1;
  float* dz1   = ws + OFF_DZ1;
  float* rinv  = ws + OFF_RINV;
  float* wbuf  = ws + OFF_W;
  float* dwbuf = ws + OFF_DW;
  float* pred  = ws + OFF_PRED;
  float* pred0 = ws + OFF_PRED0;
  float* dpre  = ws + OFF_DPRE;
  float* rcur  = ws + OFF_RCUR;
  float* rprev = ws + OFF_RPREV;
  float* r2    = ws + OFF_R2;
  float* ma    = ws + OFF_MA;
  float* va    = ws + OFF_VA;
  float* dr2g  = ws + OFF_DR2G;
  float* pr    = ws + OFF_PARTSR;
  float* pr2   = ws + OFF_PARTSR2;
  float* pb    = ws + OFF_PARTSB;
  int*   flags = (int*)(ws + OFF_FLAGS);
  const int* zg = flags + 96;   // always-zero gate (zeroed by k_init)

  dim3 b256(256);

  // init: zero r, r2, flags, loss outputs
  k_init<<<128, b256, 0, stream>>>(rcur, r2, out, flags);

  // ---------- _inf_first on X[:,0] ----------
  for (int it = 0; it < 15; it++) {
    const int* gate = flags + it;             // step-0 flag slots [0..15]
    k_sperr<<<64, b256, SMT(RD), stream>>>(rcur, Wd, X + 0 * IND, 6 * IND, e, gate);
    k_rupd<<<16, b256, SMN(IND), stream>>>(e, Wd, pred, /*use_temp=*/0, rcur, pr, gate);
    k_fin<<<1, 32, 0, stream>>>(pr, 16, pr2, 64, /*use2=*/0, flags, it, it + 1);
  }
  // spatial0 = mean_b sum_i (x0 - r@Wd^T)^2
  k_sperr<<<64, b256, SMT(RD), stream>>>(rcur, Wd, X + 0 * IND, 6 * IND, e, zg);
  k_sqred<<<512, b256, 0, stream>>>(e, (const float*)nullptr, pb);
  k_addout<<<1, 32, 0, stream>>>(out + 0, pb, 512, 1.0f / BATCH);
  // r_first output
  k_copy<<<128, b256, 0, stream>>>(out + 2 + 640, rcur);

  // ---------- time steps t = 1..5 ----------
  for (int t = 1; t <= 5; t++) {
    int fb = t * 16;
    k_stepinit<<<128, b256, 0, stream>>>(rprev, rcur, ma, va);
    // U[b,m,i] = sum_j temporal[m,i,j] * r_prev[b,j]  (hoisted out of inner loop)
    k_gemm_U<<<dim3(16, 32), b256, SMT(RD), stream>>>(rprev, Tm, U);

    for (int it = 0; it < 15; it++) {
      const int* gate = flags + fb + it;
      // hypernet forward (uses current r2)
      k_gemm_abT<<<32, b256, SMT(R2D), stream>>>(r2, W1, b1, z1, R2D, HIDD, gate);
      k_ln_fwd<<<16, b256, 0, stream>>>(z1, gg, be, h1, xhat, rinv, gate);
      k_gemm_abT<<<32, b256, SMT(HIDD), stream>>>(h1, W2, b2, z2, HIDD, HIDD, gate);
      k_gemm_abT<<<2,  b256, SMT(HIDD), stream>>>(z2, W3, b3, wbuf, HIDD, MIXD, gate);
      k_pred<<<128, b256, 0, stream>>>(U, wbuf, pred, pred0, (it == 0) ? 1 : 0, gate);
      // spatial residual (uses current r)
      k_sperr<<<64, b256, SMT(RD), stream>>>(rcur, Wd, X + t * IND, 6 * IND, e, gate);
      // backward to r2
      k_dpre<<<128, b256, 0, stream>>>(pred, rcur, dpre, gate);
      k_dw<<<512, b256, 0, stream>>>(dpre, U, dwbuf, gate);
      k_gemm_abN<<<32, b256, SMN(MIXD), stream>>>(dwbuf, W3, dz2, MIXD, HIDD, gate);
      k_gemm_abN<<<32, b256, SMN(HIDD), stream>>>(dz2, W2, dh1, HIDD, HIDD, gate);
      k_ln_bwd<<<16, b256, 0, stream>>>(dh1, h1, xhat, rinv, gg, dz1, gate);
      k_gemm_abN<<<8, b256, SMN(HIDD), stream>>>(dz1, W1, dr2g, HIDD, R2D, gate);
      // updates (both read pre-update state)
      k_rupd<<<16, b256, SMN(IND), stream>>>(e, Wd, pred, /*use_temp=*/1, rcur, pr, gate);
      float p1 = (float)(1.0 - pow(0.9, (double)(it + 1)));
      float p2 = (float)(1.0 - pow(0.999, (double)(it + 1)));
      k_adam<<<64, b256, 0, stream>>>(dr2g, r2, ma, va, 1.0f / p1, 1.0f / p2, pr2, gate);
      k_fin<<<1, 32, 0, stream>>>(pr, 16, pr2, 64, /*use2=*/1, flags, fb + it, fb + it + 1);
    }

    // sl_t = mean_b sum_i (x_t - r@Wd^T)^2 with final r
    k_sperr<<<64, b256, SMT(RD), stream>>>(rcur, Wd, X + t * IND, 6 * IND, e, zg);
    k_sqred<<<512, b256, 0, stream>>>(e, (const float*)nullptr, pb);
    k_addout<<<1, 32, 0, stream>>>(out + 0, pb, 512, 1.0f / BATCH);
    // tl_t: hypernet forward with FINAL r2 -> pred_f
    k_gemm_abT<<<32, b256, SMT(R2D), stream>>>(r2, W1, b1, z1, R2D, HIDD, zg);
    k_ln_fwd<<<16, b256, 0, stream>>>(z1, gg, be, h1, xhat, rinv, zg);
    k_gemm_abT<<<32, b256, SMT(HIDD), stream>>>(h1, W2, b2, z2, HIDD, HIDD, zg);
    k_gemm_abT<<<2,  b256, SMT(HIDD), stream>>>(z2, W3, b3, wbuf, HIDD, MIXD, zg);
    k_pred<<<128, b256, 0, stream>>>(U, wbuf, pred, pred0, 0, zg);
    k_sqred<<<128, b256, 0, stream>>>(rcur, pred, pb);
    k_addout<<<1, 32, 0, stream>>>(out + 1, pb, 128, 1.0f / BATCH);
    // r2_losses[:, t-1] vs pred at ENTRY r2 (== iteration-0 prediction, saved as pred0)
    k_r2loss<<<16, b256, 0, stream>>>(rcur, pred0, out + 2, t - 1);
  }

  // final r2 output
  k_copy<<<64, b256, 0, stream>>>(out + 2 + 640 + 32768, r2);
}